// MACE_17935783428300
// MI455X (gfx1250) — compile-verified
//
#include <hip/hip_runtime.h>
#include <math.h>

typedef float v2f __attribute__((ext_vector_type(2)));
typedef float v8f __attribute__((ext_vector_type(8)));

#define FDIM 64
#define NRAD 8
#define HIDR 64
#define RCOLS 512          // F*4*2
#define RMAXC 5.0f
#define INV_SQRT_AVG 0.25f // 1/sqrt(16)
#define PI_F 3.14159265358979323846f
#define SNUM 10

__device__ __forceinline__ float silu_f(float x) { return x / (1.0f + expf(-x)); }

// D = A(16x4,f32) * B(4x16,f32) + C ; f32 WMMA (wave32)
__device__ __forceinline__ v8f wmma4(v2f a, v2f b, v8f c) {
    return __builtin_amdgcn_wmma_f32_16x16x4_f32(false, a, false, b, (short)0, c,
                                                 false, false);
}

__device__ __forceinline__ int l_lo(int l) { return (l == 0) ? 0 : (l == 1) ? 1 : (l == 2) ? 4 : 9; }
__device__ __forceinline__ int l_hi(int l) { return (l == 0) ? 1 : (l == 1) ? 4 : (l == 2) ? 9 : 16; }

// ---------------------------------------------------------------- edge geometry
__global__ void k_edge_geom(const float* __restrict__ vec, float* __restrict__ Y,
                            float* __restrict__ Remb, int E) {
    int e = blockIdx.x * blockDim.x + threadIdx.x;
    if (e >= E) return;
    float vx = vec[e * 3 + 0], vy = vec[e * 3 + 1], vz = vec[e * 3 + 2];
    float r = sqrtf(vx * vx + vy * vy + vz * vz + 1e-12f);
    float inv = 1.0f / r;
    float x = vx * inv, y = vy * inv, z = vz * inv;
    float r2 = x * x + y * y + z * z;
    float* Yr = Y + (size_t)e * 16;
    Yr[0] = 1.0f;
    Yr[1] = x;  Yr[2] = y;  Yr[3] = z;
    Yr[4] = x * y;  Yr[5] = y * z;  Yr[6] = 3.0f * z * z - r2;
    Yr[7] = x * z;  Yr[8] = x * x - y * y;
    Yr[9]  = y * (3.0f * x * x - y * y);
    Yr[10] = x * y * z;
    Yr[11] = y * (5.0f * z * z - r2);
    Yr[12] = z * (5.0f * z * z - 3.0f * r2);
    Yr[13] = x * (5.0f * z * z - r2);
    Yr[14] = z * (x * x - y * y);
    Yr[15] = x * (x * x - 3.0f * y * y);

    float u = r / RMAXC;
    float env = (u < 1.0f) ? (1.0f - u) * (1.0f - u) * (1.0f + 2.0f * u) : 0.0f;
    float base = env / (u + 1e-6f);
    float* Rr = Remb + (size_t)e * 8;
#pragma unroll
    for (int k = 0; k < 8; ++k) Rr[k] = sinf(PI_F * (float)(k + 1) * u) * base;
}

// ---------------------------------------------------------------- feats init / zero
__global__ void k_init_feats(const int* __restrict__ spec, const float* __restrict__ wemb,
                             float* __restrict__ feats, size_t total) {
    size_t idx = (size_t)blockIdx.x * blockDim.x + threadIdx.x;
    if (idx >= total) return;
    int n = (int)(idx >> 10);
    int rem = (int)(idx & 1023);
    int f = rem >> 4, m = rem & 15;
    feats[idx] = (m == 0) ? wemb[spec[n] * FDIM + f] : 0.0f;
}

__global__ void k_zero(float* __restrict__ p, size_t n) {
    size_t i = (size_t)blockIdx.x * blockDim.x + threadIdx.x;
    if (i < n) p[i] = 0.0f;
}

// ---------------------------------------------------------------- fused radial MLP +
// message + scatter.  128 threads = 4 waves, 16 edges per block.
// L0: layer-0 specialization (feats columns m>0 are exactly zero -> gather only s0).
template <bool L0>
__global__ __launch_bounds__(128) void k_edge_message(
    const float* __restrict__ Yg, const float* __restrict__ Rembg,
    const float* __restrict__ w1, const float* __restrict__ w2,
    const float* __restrict__ feats, float* __restrict__ agg,
    const int* __restrict__ senders, const int* __restrict__ receivers, int E) {
    __shared__ float sRemb[16][8];     // A for GEMM1 (16 edges x 8)
    __shared__ float sH[16][64];       // silu(Remb @ w1)
    __shared__ float sR[16][512];      // R = H @ w2  (32 KB)
    __shared__ int sSend[16], sRecv[16];

    const int tid = threadIdx.x;
    const int wid = tid >> 5;          // wave 0..3
    const int lane = tid & 31;
    const int lhalf = lane & 15;
    const int koff = (lane < 16) ? 0 : 2;   // K split across lane halves (ISA 7.12.2)
    const int e0 = blockIdx.x * 16;

    // stage Remb tile and edge indices
    {
        int e = tid >> 3, k = tid & 7;
        int eg = e0 + e;
        sRemb[e][k] = (eg < E) ? Rembg[(size_t)eg * 8 + k] : 0.0f;
    }
    if (tid < 16) {
        int eg = e0 + tid;
        sSend[tid] = (eg < E) ? senders[eg] : 0;
        sRecv[tid] = (eg < E) ? receivers[eg] : 0;
    }
    __syncthreads();

    // GEMM1: H(16x64) = silu(Remb(16x8) @ w1(8x64)); wave wid owns col tile [16*wid,16*wid+16)
    {
        v8f acc = {0, 0, 0, 0, 0, 0, 0, 0};
        const int col = wid * 16 + lhalf;
#pragma unroll
        for (int kc = 0; kc < 2; ++kc) {
            v2f a, b;
            a[0] = sRemb[lhalf][kc * 4 + koff + 0];
            a[1] = sRemb[lhalf][kc * 4 + koff + 1];
            b[0] = w1[(kc * 4 + koff + 0) * HIDR + col];
            b[1] = w1[(kc * 4 + koff + 1) * HIDR + col];
            acc = wmma4(a, b, acc);
        }
#pragma unroll
        for (int j = 0; j < 8; ++j) {
            int m = j + ((lane < 16) ? 0 : 8);
            sH[m][col] = silu_f(acc[j]);
        }
    }
    __syncthreads();

    // GEMM2: R(16x512) = H(16x64) @ w2(64x512); wave wid does tiles wid, wid+4, ...
#pragma unroll 1
    for (int t = 0; t < 8; ++t) {
        const int nt = wid + 4 * t;
        const int col = nt * 16 + lhalf;
        v8f acc = {0, 0, 0, 0, 0, 0, 0, 0};
#pragma unroll
        for (int kc = 0; kc < 16; ++kc) {
            v2f a, b;
            a[0] = sH[lhalf][kc * 4 + koff + 0];
            a[1] = sH[lhalf][kc * 4 + koff + 1];
            b[0] = w2[(kc * 4 + koff + 0) * RCOLS + col];
            b[1] = w2[(kc * 4 + koff + 1) * RCOLS + col];
            acc = wmma4(a, b, acc);
        }
#pragma unroll
        for (int j = 0; j < 8; ++j) {
            int m = j + ((lane < 16) ? 0 : 8);
            sR[m][col] = acc[j];
        }
    }
    __syncthreads();

    // messages + atomic scatter; coalesced float4 gather of sender features
    for (int e = 0; e < 16; ++e) {
        int eg = e0 + e;
        if (eg >= E) break;
        const int snd = sSend[e];
        const int rcv = sRecv[e];
        const float* fsrow = feats + (size_t)snd * 1024;
        const float4* fs4 = reinterpret_cast<const float4*>(fsrow);
        const float4* Y4 = reinterpret_cast<const float4*>(Yg + (size_t)eg * 16);
        float* aggrow = agg + (size_t)rcv * 1024;
        const float* Re = sR[e];
#pragma unroll
        for (int it = 0; it < 2; ++it) {
            const int v = it * 128 + tid;   // vec4 index 0..255 over F*16
            const int f = v >> 2;
            const int mbase = (v & 3) * 4;
            float4 yv = Y4[v & 3];
            float ya[4] = {yv.x, yv.y, yv.z, yv.w};
            float s0 = fsrow[f * 16];
            float fsa[4];
            if (L0) {
                // layer 0: fs[:, m>0] == 0, fs[:, 0] == s0
                fsa[0] = (mbase == 0) ? s0 : 0.0f;
                fsa[1] = 0.0f; fsa[2] = 0.0f; fsa[3] = 0.0f;
            } else {
                float4 fv = fs4[v];
                fsa[0] = fv.x; fsa[1] = fv.y; fsa[2] = fv.z; fsa[3] = fv.w;
            }
#pragma unroll
            for (int c = 0; c < 4; ++c) {
                int m = mbase + c;
                int l = (m == 0) ? 0 : (m < 4) ? 1 : (m < 9) ? 2 : 3;
                float R0 = Re[f * 8 + l * 2 + 0];
                float R1 = Re[f * 8 + l * 2 + 1];
                float msg = (R0 * s0 * ya[c] + R1 * fsa[c]) * INV_SQRT_AVG;
                __hip_atomic_fetch_add(&aggrow[v * 4 + c], msg, __ATOMIC_RELAXED,
                                       __HIP_MEMORY_SCOPE_AGENT);
            }
        }
    }
}

// ---------------------------------------------------------------- layer-0 node update:
// h = species_linear(agg, W0[spec]); feats = h*poly(h[:,0]); out0 = feats[:,0]@w_read0
__global__ __launch_bounds__(128) void k_node0(
    const float* __restrict__ agg, const int* __restrict__ spec,
    const float* __restrict__ wskip, const float* __restrict__ wpoly,
    const float* __restrict__ wread0, float* __restrict__ feats,
    float* __restrict__ out, int N) {
    __shared__ float sB[64][16];   // agg[n]
    __shared__ float sD[64][16];   // species_linear output
    __shared__ float sred[64];

    const int n = blockIdx.x;
    const int tid = threadIdx.x;
    const int wid = tid >> 5;      // wave = l index (0..3)
    const int lane = tid & 31;
    const int lhalf = lane & 31 & 15;
    const int koff = (lane < 16) ? 0 : 2;
    const int sp = spec[n];

    const float* an = agg + (size_t)n * 1024;
    for (int i = tid; i < 1024; i += 128) sB[i >> 4][i & 15] = an[i];
    __syncthreads();

    // wave l: D = W_l^T(64x64) @ B(64x16); columns are independent, so compute all
    // 16 and commit only columns in this l's slice (disjoint across waves).
    const float* Wl = wskip + ((size_t)sp * 4 + wid) * 4096;  // layer 0
    const int lo = l_lo(wid), hi = l_hi(wid);
#pragma unroll 1
    for (int mt = 0; mt < 4; ++mt) {
        const int g = mt * 16 + lhalf;
        v8f acc = {0, 0, 0, 0, 0, 0, 0, 0};
#pragma unroll
        for (int kc = 0; kc < 16; ++kc) {
            v2f a, b;
            a[0] = Wl[(kc * 4 + koff + 0) * 64 + g];   // A[g][f] = W[f][g]
            a[1] = Wl[(kc * 4 + koff + 1) * 64 + g];
            b[0] = sB[kc * 4 + koff + 0][lhalf];
            b[1] = sB[kc * 4 + koff + 1][lhalf];
            acc = wmma4(a, b, acc);
        }
#pragma unroll
        for (int j = 0; j < 8; ++j) {
            int row = mt * 16 + j + ((lane < 16) ? 0 : 8);
            if (lhalf >= lo && lhalf < hi) sD[row][lhalf] = acc[j];
        }
    }
    __syncthreads();

    if (tid < 64) {
        const int g = tid;
        float s = sD[g][0];
        float c0 = wpoly[(0 * SNUM + sp) * FDIM + g];
        float c1 = wpoly[(1 * SNUM + sp) * FDIM + g];
        float c2 = wpoly[(2 * SNUM + sp) * FDIM + g];
        float scale = c0 + c1 * s + c2 * s * s;
        float* fn = feats + (size_t)n * 1024 + g * 16;
#pragma unroll
        for (int m = 0; m < 16; ++m) fn[m] = sD[g][m] * scale;
        sred[g] = sD[g][0] * scale * wread0[g];
    }
    __syncthreads();
    if (tid == 0) {
        float acc = 0.0f;
        for (int g = 0; g < 64; ++g) acc += sred[g];
        out[(size_t)n * 2 + 0] = acc;
    }
}

// ---------------------------------------------------------------- layer-1 node update:
// only column 0 feeds the readout -> matvec is the right shape (WMMA would be 1/16 used)
__global__ __launch_bounds__(128) void k_node1(
    const float* __restrict__ agg, const float* __restrict__ feats,
    const int* __restrict__ spec, const float* __restrict__ wskip,
    const float* __restrict__ wpoly, const float* __restrict__ wmlp1,
    const float* __restrict__ wmlp2, float* __restrict__ out, int N) {
    __shared__ float sF0[64], sH0[64], sP[64], sM[32];
    const int n = blockIdx.x;
    const int tid = threadIdx.x;
    const int sp = spec[n];

    if (tid < 64) {
        sF0[tid] = feats[(size_t)n * 1024 + tid * 16];
        sH0[tid] = agg[(size_t)n * 1024 + tid * 16];
    }
    __syncthreads();
    if (tid < 64) {
        const int g = tid;
        const float* W = wskip + ((size_t)(SNUM + sp) * 4 + 0) * 4096;  // layer 1, l=0
        float sc = 0.0f;
        for (int f = 0; f < 64; ++f) sc += sF0[f] * W[f * 64 + g];
        float s = sH0[g];
        float c0 = wpoly[((3 + 0) * SNUM + sp) * FDIM + g];
        float c1 = wpoly[((3 + 1) * SNUM + sp) * FDIM + g];
        float c2 = wpoly[((3 + 2) * SNUM + sp) * FDIM + g];
        float scale = c0 + c1 * s + c2 * s * s;
        sP[g] = s * scale + sc;
    }
    __syncthreads();
    if (tid < 32) {
        float h = 0.0f;
        for (int g = 0; g < 64; ++g) h += sP[g] * wmlp1[g * 32 + tid];
        sM[tid] = silu_f(h) * wmlp2[tid];
    }
    __syncthreads();
    if (tid == 0) {
        float acc = 0.0f;
        for (int t = 0; t < 32; ++t) acc += sM[t];
        out[(size_t)n * 2 + 1] = acc;
    }
}

// ----------------------------------------------------------------
extern "C" void kernel_launch(void* const* d_in, const int* in_sizes, int n_in,
                              void* d_out, int out_size, void* d_ws, size_t ws_size,
                              hipStream_t stream) {
    const float* vectors = (const float*)d_in[0];
    const int* spec      = (const int*)d_in[1];
    const int* senders   = (const int*)d_in[2];
    const int* receivers = (const int*)d_in[3];
    const float* w_embed = (const float*)d_in[4];
    const float* w_r1    = (const float*)d_in[5];
    const float* w_r2    = (const float*)d_in[6];
    const float* w_skip  = (const float*)d_in[7];
    const float* w_poly  = (const float*)d_in[8];
    const float* w_read0 = (const float*)d_in[9];
    const float* w_mlp1  = (const float*)d_in[10];
    const float* w_mlp2  = (const float*)d_in[11];
    float* out = (float*)d_out;

    const int E = in_sizes[0] / 3;
    const int N = in_sizes[1];

    float* ws    = (float*)d_ws;
    float* Y     = ws;                           // E*16
    float* Remb  = Y + (size_t)E * 16;           // E*8
    float* feats = Remb + (size_t)E * 8;         // N*1024
    float* agg   = feats + (size_t)N * 1024;     // N*1024

    const size_t nodeElems = (size_t)N * 1024;
    const unsigned zb = (unsigned)((nodeElems + 255) / 256);
    const unsigned eb = (unsigned)((E + 15) / 16);

    k_edge_geom<<<(E + 255) / 256, 256, 0, stream>>>(vectors, Y, Remb, E);
    k_init_feats<<<zb, 256, 0, stream>>>(spec, w_embed, feats, nodeElems);

    // ---- layer 0 (sparse feats: gather only s0)
    k_zero<<<zb, 256, 0, stream>>>(agg, nodeElems);
    k_edge_message<true><<<eb, 128, 0, stream>>>(Y, Remb, w_r1, w_r2, feats, agg,
                                                 senders, receivers, E);
    k_node0<<<N, 128, 0, stream>>>(agg, spec, w_skip, w_poly, w_read0, feats, out, N);

    // ---- layer 1 (dense feats)
    k_zero<<<zb, 256, 0, stream>>>(agg, nodeElems);
    k_edge_message<false><<<eb, 128, 0, stream>>>(Y, Remb, w_r1 + NRAD * HIDR,
                                                  w_r2 + HIDR * RCOLS, feats, agg,
                                                  senders, receivers, E);
    k_node1<<<N, 128, 0, stream>>>(agg, feats, spec, w_skip, w_poly, w_mlp1, w_mlp2,
                                   out, N);
}